// AdjGCN_23596550324896
// MI455X (gfx1250) — compile-verified
//
#include <hip/hip_runtime.h>
#include <math.h>

typedef __attribute__((ext_vector_type(2))) float v2f;
typedef __attribute__((ext_vector_type(8))) float v8f;

// ---------------------------------------------------------------------------
// GEMM: H[n x d_store] = X[n x 128] @ W[128 x d_w]  (d_w <= DO, DO multiple of 16)
// fp32 WMMA 16x16x4, K=128 -> 32 K-steps.
// Layout per CDNA5 ISA 7.12.2:
//   A 16x4 f32 : lanes 0-15 row M=lane, VGPR0=K, VGPR1=K+1; lanes 16-31: K+2,K+3
//   B 4x16 f32 : lanes 0-15 col N=lane, VGPR0=K, VGPR1=K+1; lanes 16-31: K+2,K+3
//   C/D 16x16  : VGPR j -> M=j (lanes 0-15) / M=j+8 (lanes 16-31), N=lane&15
// ---------------------------------------------------------------------------
template <int DO>
__global__ __launch_bounds__(256) void gcn_gemm_wmma(
    const float* __restrict__ X, const float* __restrict__ Wm,
    float* __restrict__ H, int n, int d_w, int d_store)
{
    __shared__ float Wlds[128 * DO];
    for (int i = threadIdx.x; i < 128 * DO; i += 256) {
        int row = i / DO, col = i - row * DO;
        Wlds[i] = (col < d_w) ? Wm[row * d_w + col] : 0.0f;
    }
    __syncthreads();

    const int wave = threadIdx.x >> 5;
    const int lane = threadIdx.x & 31;
    const int row0 = blockIdx.x * 128 + wave * 16;
    if (row0 + 16 > n) return;

    const int rlo  = lane & 15;        // A row / B,C column within tile
    const int koff = (lane >> 4) * 2;  // upper half-wave carries K+2,K+3

    // Preload all A fragments for this 16-row tile (K = 128)
    const float* xr = X + (long long)(row0 + rlo) * 128;
    v2f a[32];
#pragma unroll
    for (int ks = 0; ks < 32; ++ks) {
        int k = ks * 4 + koff;
        v2f t; t.x = xr[k]; t.y = xr[k + 1];
        a[ks] = t;
    }

    const int mbase = (lane >> 4) * 8;
#pragma unroll 1
    for (int ct = 0; ct < DO / 16; ++ct) {
        const int cc = ct * 16 + rlo;
        v8f c = {};
#pragma unroll
        for (int ks = 0; ks < 32; ++ks) {
            int k = ks * 4 + koff;
            v2f b; b.x = Wlds[k * DO + cc]; b.y = Wlds[(k + 1) * DO + cc];
            c = __builtin_amdgcn_wmma_f32_16x16x4_f32(
                    false, a[ks], false, b, (short)0, c, false, false);
        }
        if (cc < d_store) {
#pragma unroll
            for (int j = 0; j < 8; ++j)
                H[(long long)(row0 + mbase + j) * d_store + cc] = c[j];
        }
    }
}

// ---------------------------------------------------------------------------
// Edge scatter-add: AGG[dst[e]] += H[src[e]] * w[e].  One wave per edge,
// float4 gather, f32 atomic adds (L2-resident: N*128*4B = 51 MB < 192 MB L2).
// ---------------------------------------------------------------------------
template <int D>
__global__ __launch_bounds__(256) void gcn_scatter_add(
    const float* __restrict__ H, const long long* __restrict__ src,
    const long long* __restrict__ dst, const float* __restrict__ ew,
    float* __restrict__ AGG, int E)
{
    const int lane = threadIdx.x & 31;
    const int e = blockIdx.x * 8 + (threadIdx.x >> 5);
    if (e >= E) return;
    const long long s = src[e];
    const long long d = dst[e];
    const float w = ew[e];
    const float4* hs = (const float4*)(H + s * D);
    float* ag = AGG + d * D;
    for (int i = lane; i < D / 4; i += 32) {
        float4 v = hs[i];
        __hip_atomic_fetch_add(ag + i * 4 + 0, v.x * w, __ATOMIC_RELAXED, __HIP_MEMORY_SCOPE_AGENT);
        __hip_atomic_fetch_add(ag + i * 4 + 1, v.y * w, __ATOMIC_RELAXED, __HIP_MEMORY_SCOPE_AGENT);
        __hip_atomic_fetch_add(ag + i * 4 + 2, v.z * w, __ATOMIC_RELAXED, __HIP_MEMORY_SCOPE_AGENT);
        __hip_atomic_fetch_add(ag + i * 4 + 3, v.w * w, __ATOMIC_RELAXED, __HIP_MEMORY_SCOPE_AGENT);
    }
}

// ---------------------------------------------------------------------------
// O[i] = relu(A[i] + b[i mod 128])
// ---------------------------------------------------------------------------
__global__ __launch_bounds__(256) void gcn_bias_relu(
    const float* __restrict__ A, const float* __restrict__ b,
    float* __restrict__ O, long long total)
{
    long long i = (long long)blockIdx.x * blockDim.x + threadIdx.x;
    if (i >= total) return;
    float v = A[i] + b[(int)(i & 127)];
    O[i] = v > 0.0f ? v : 0.0f;
}

// ---------------------------------------------------------------------------
// Final: out = log_softmax(A + b3) over 40 classes. One wave32 per node;
// lanes cover cols 0..31 plus 32..39 on lanes 0..7; shuffle reductions.
// ---------------------------------------------------------------------------
__global__ __launch_bounds__(256) void gcn_logsoftmax40(
    const float* __restrict__ A, const float* __restrict__ b,
    float* __restrict__ O, int n)
{
    const int lane = threadIdx.x & 31;
    const int node = blockIdx.x * 8 + (threadIdx.x >> 5);
    if (node >= n) return;
    const float* z = A + (long long)node * 40;
    float a0 = z[lane] + b[lane];
    float a1 = (lane < 8) ? (z[32 + lane] + b[32 + lane]) : -INFINITY;
    float m = fmaxf(a0, a1);
    for (int off = 16; off; off >>= 1) m = fmaxf(m, __shfl_xor(m, off, 32));
    float s = expf(a0 - m) + ((lane < 8) ? expf(a1 - m) : 0.0f);
    for (int off = 16; off; off >>= 1) s += __shfl_xor(s, off, 32);
    float ls = m + logf(s);
    float* o = O + (long long)node * 40;
    o[lane] = a0 - ls;
    if (lane < 8) o[32 + lane] = a1 - ls;
}

// ---------------------------------------------------------------------------
extern "C" void kernel_launch(void* const* d_in, const int* in_sizes, int n_in,
                              void* d_out, int out_size, void* d_ws, size_t ws_size,
                              hipStream_t stream)
{
    const float*     x  = (const float*)d_in[0];
    const long long* ei = (const long long*)d_in[1];
    const float*     ew = (const float*)d_in[2];
    const float*     W1 = (const float*)d_in[3];
    const float*     b1 = (const float*)d_in[4];
    const float*     W2 = (const float*)d_in[5];
    const float*     b2 = (const float*)d_in[6];
    const float*     W3 = (const float*)d_in[7];
    const float*     b3 = (const float*)d_in[8];

    const int n = in_sizes[0] / 128;
    const int E = in_sizes[1] / 2;
    const long long* src = ei;
    const long long* dst = ei + E;

    float* bufH = (float*)d_ws;                  // n*128 floats
    float* bufA = bufH + (size_t)n * 128;        // n*128 floats
    float* out  = (float*)d_out;

    const dim3 blk(256);
    const int gGemm = (n + 127) / 128;
    const int gScat = (E + 7) / 8;
    const long long tot128 = (long long)n * 128;
    const int gEw = (int)((tot128 + 255) / 256);
    const int gSm = (n + 7) / 8;

    // ---- Layer 1: H1 = x@W1 -> AGG1 -> relu(+b1) ----
    gcn_gemm_wmma<128><<<gGemm, blk, 0, stream>>>(x, W1, bufH, n, 128, 128);
    hipMemsetAsync(bufA, 0, (size_t)n * 128 * sizeof(float), stream);
    gcn_scatter_add<128><<<gScat, blk, 0, stream>>>(bufH, src, dst, ew, bufA, E);
    gcn_bias_relu<<<gEw, blk, 0, stream>>>(bufA, b1, bufH, tot128);

    // ---- Layer 2 ----
    gcn_gemm_wmma<128><<<gGemm, blk, 0, stream>>>(bufH, W2, bufA, n, 128, 128);
    hipMemsetAsync(bufH, 0, (size_t)n * 128 * sizeof(float), stream);
    gcn_scatter_add<128><<<gScat, blk, 0, stream>>>(bufA, src, dst, ew, bufH, E);
    gcn_bias_relu<<<gEw, blk, 0, stream>>>(bufH, b2, bufA, tot128);

    // ---- Layer 3: 128 -> 40 (W padded to 48 cols in LDS), log_softmax ----
    gcn_gemm_wmma<48><<<gGemm, blk, 0, stream>>>(bufA, W3, bufH, n, 40, 40);
    hipMemsetAsync(bufA, 0, (size_t)n * 40 * sizeof(float), stream);
    gcn_scatter_add<40><<<gScat, blk, 0, stream>>>(bufH, src, dst, ew, bufA, E);
    gcn_logsoftmax40<<<gSm, blk, 0, stream>>>(bufA, b3, out, n);
}